// EquivLayerNormJIT_4114578670339
// MI455X (gfx1250) — compile-verified
//
#include <hip/hip_runtime.h>

// ---------------------------------------------------------------------------
// EquivLayerNorm, irreps = 128x0e + 64x1o + 32x2e  (F = 480, rows = 131072)
// Bandwidth-bound (503 MB @ 23.3 TB/s ~ 21.6 us). No matmul content -> no
// WMMA; instead use gfx1250 async global->LDS DMA (ASYNCcnt) to double-buffer
// rows per wave, wave32 __shfl_xor butterflies for the reductions.
// All CDNA5-specific ops issued via inline asm with integer operands (no
// address-space pointer casts needed).
// ---------------------------------------------------------------------------

#define F_DIM   480
#define WPB     8      // waves per block (256 threads)
#define RPW     4      // rows per wave (pipeline over LDS double buffer)
#define EPS_LN  1e-5f

// ---- gfx1250 async copy + wait-counter primitives (inline asm) ------------
// vdst VGPR holds the per-lane LDS byte address; vaddr is a 64-bit VGPR pair.
#define ASYNC_LOAD_B128(gaddr_u64, ldsoff_u32)                                 \
  asm volatile("global_load_async_to_lds_b128 %0, %1, off" ::                  \
                   "v"((unsigned)(ldsoff_u32)),                                \
                   "v"((unsigned long long)(gaddr_u64))                        \
               : "memory")

#define WAIT_ASYNC(n) asm volatile("s_wait_asynccnt %0" ::"i"(n) : "memory")
#define WAIT_DS0()    asm volatile("s_wait_dscnt 0" ::: "memory")

// ---- wave32 butterfly all-reduce ------------------------------------------
__device__ __forceinline__ float wave_sum(float x) {
  x += __shfl_xor(x, 16, 32);
  x += __shfl_xor(x, 8, 32);
  x += __shfl_xor(x, 4, 32);
  x += __shfl_xor(x, 2, 32);
  x += __shfl_xor(x, 1, 32);
  return x;
}

// Issue async DMA of one 1920-byte row into this wave's LDS buffer.
// 3 full b128 chunks (32 lanes x 16B = 512B each) + one 24-lane tail chunk.
__device__ __forceinline__ void issue_row(const float *__restrict__ src,
                                          unsigned lds_off, int lane) {
  const unsigned long long g =
      (unsigned long long)(size_t)(src) + (unsigned)lane * 16u;
  const unsigned l = lds_off + (unsigned)lane * 16u;
  ASYNC_LOAD_B128(g + 0 * 512, l + 0 * 512);
  ASYNC_LOAD_B128(g + 1 * 512, l + 1 * 512);
  ASYNC_LOAD_B128(g + 2 * 512, l + 2 * 512);
  if (lane < 24) {
    ASYNC_LOAD_B128(g + 3 * 512, l + 3 * 512);
  }
}

__device__ __forceinline__ void
process_row(const float *tr,          // LDS row (480 floats)
            float *__restrict__ orow, // global out row
            int lane, float s0, float s1, float s2, const float *off_s) {
  // j*32+lane layout: j<4 -> scalars [0,128), j<10 -> vectors [128,320),
  // else l=2 [320,480). Boundaries align with j since 128,320 are mult. of 32.
  float v[15];
#pragma unroll
  for (int j = 0; j < 15; ++j)
    v[j] = tr[j * 32 + lane];

  // scalar mean over the 128 l=0 copies
  float mean = wave_sum(v[0] + v[1] + v[2] + v[3]) * (1.0f / 128.0f);

  float c[4];
  float ss0 = 0.f, ss1 = 0.f, ss2 = 0.f;
#pragma unroll
  for (int j = 0; j < 4; ++j) {
    c[j] = v[j] - mean;
    ss0 += c[j] * c[j];
  }
#pragma unroll
  for (int j = 4; j < 10; ++j)
    ss1 += v[j] * v[j];
#pragma unroll
  for (int j = 10; j < 15; ++j)
    ss2 += v[j] * v[j];

  ss0 = wave_sum(ss0);
  ss1 = wave_sum(ss1);
  ss2 = wave_sum(ss2);

  const float i0 = rsqrtf(ss0 * (1.0f / 127.0f) + EPS_LN) * s0;
  const float i1 = rsqrtf(ss1 * (1.0f / 63.0f) + EPS_LN) * s1;
  const float i2 = rsqrtf(ss2 * (1.0f / 31.0f) + EPS_LN) * s2;

#pragma unroll
  for (int j = 0; j < 4; ++j)
    orow[j * 32 + lane] = c[j] * i0 + off_s[j]; // offset only on scalars
#pragma unroll
  for (int j = 4; j < 10; ++j)
    orow[j * 32 + lane] = v[j] * i1;
#pragma unroll
  for (int j = 10; j < 15; ++j)
    orow[j * 32 + lane] = v[j] * i2;

  // keep the LDS reads ordered before the next async issue into this buffer
  asm volatile("" ::: "memory");
}

__global__ __launch_bounds__(WPB * 32) void
equiv_layernorm_kernel(const float *__restrict__ feat,
                       const float *__restrict__ scale,
                       const float *__restrict__ offset,
                       float *__restrict__ out, int nrows) {
  __shared__ __align__(16) float tile[WPB][2][F_DIM]; // 30720 B / block

  const int lane = threadIdx.x & 31;
  const int wave = threadIdx.x >> 5;

  const float s0 = scale[0], s1 = scale[1], s2 = scale[2];
  float off_s[4];
#pragma unroll
  for (int j = 0; j < 4; ++j)
    off_s[j] = offset[j * 32 + lane];

  // Low 32 bits of a generic pointer to __shared__ = LDS byte offset
  // (flat shared address = {SHARED_BASE[hi32], lds_offset[lo32]}).
  const unsigned lds_base = (unsigned)(size_t)(&tile[wave][0][0]);
  const unsigned lds_buf[2] = {lds_base, lds_base + F_DIM * 4u};

  const int row0 = (blockIdx.x * WPB + wave) * RPW;
  if (row0 >= nrows)
    return;

  issue_row(feat + (size_t)row0 * F_DIM, lds_buf[0], lane);

  for (int r = 0; r < RPW; ++r) {
    const int row = row0 + r;
    if (row >= nrows)
      break;
    const bool have_next = (r + 1 < RPW) && (row + 1 < nrows);
    if (have_next) {
      WAIT_DS0(); // buffer (r+1)&1 was read at iteration r-1; guard WAR
      issue_row(feat + (size_t)(row + 1) * F_DIM, lds_buf[(r + 1) & 1], lane);
      WAIT_ASYNC(4); // 4 in flight for row+1; row r's 4 chunks are complete
    } else {
      WAIT_ASYNC(0);
    }
    process_row(&tile[wave][r & 1][0], out + (size_t)row * F_DIM, lane, s0, s1,
                s2, off_s);
  }
}

extern "C" void kernel_launch(void *const *d_in, const int *in_sizes, int n_in,
                              void *d_out, int out_size, void *d_ws,
                              size_t ws_size, hipStream_t stream) {
  const float *feat = (const float *)d_in[0];   // (131072, 480) f32
  const float *scale = (const float *)d_in[1];  // (1, 3) f32
  const float *offset = (const float *)d_in[2]; // (1, 480) f32
  float *out = (float *)d_out;

  const int nrows = in_sizes[0] / F_DIM;
  const int rows_per_block = WPB * RPW;
  const int blocks = (nrows + rows_per_block - 1) / rows_per_block;

  equiv_layernorm_kernel<<<blocks, WPB * 32, 0, stream>>>(feat, scale, offset,
                                                          out, nrows);
}